// GCN_1967095022252
// MI455X (gfx1250) — compile-verified
//
#include <hip/hip_runtime.h>
#include <math.h>

typedef float v2f __attribute__((ext_vector_type(2)));
typedef float v8f __attribute__((ext_vector_type(8)));

#define NEG_SLOPE 0.01f
#define TPB 256

// ---------------- degree / normalization ----------------
__global__ void deg_init_kernel(float* deg, int n) {
  int i = blockIdx.x * blockDim.x + threadIdx.x;
  if (i < n) deg[i] = 1.0f;  // self-loop contributes 1 to every target degree
}

__global__ void deg_count_kernel(const long long* __restrict__ col, float* deg, int E) {
  int e = blockIdx.x * blockDim.x + threadIdx.x;
  if (e < E) atomicAdd(&deg[(int)col[e]], 1.0f);
}

__global__ void dinv_kernel(float* deg, int n) {
  int i = blockIdx.x * blockDim.x + threadIdx.x;
  if (i < n) deg[i] = rsqrtf(deg[i]);  // deg >= 1 always (self-loops)
}

// ---------------- padding passes (make GEMM1 inner loop branch-free) ----------------
// x [N,10] -> xp [N,16] (cols 10..15 = 0); rows 64B aligned.
__global__ void pad_x_kernel(const float* __restrict__ x, float* __restrict__ xp, int n) {
  int tid = blockIdx.x * blockDim.x + threadIdx.x;
  if (tid >= n * 16) return;
  int node = tid >> 4;
  int f    = tid & 15;
  xp[tid] = (f < 10) ? x[(size_t)node * 10 + f] : 0.0f;
}

// W1 [10,32] -> W1p [12,32] (rows 10,11 = 0)
__global__ void pad_w1_kernel(const float* __restrict__ w, float* __restrict__ wp) {
  int tid = blockIdx.x * blockDim.x + threadIdx.x;  // 384 threads
  if (tid >= 12 * 32) return;
  int k = tid >> 5;
  wp[tid] = (k < 10) ? w[tid] : 0.0f;
}

// ---------------- dense GEMM 1: xw1 = xp @ W1p  ([N,16(K=12)] @ [12,32]) ----------------
// One wave -> 16 output rows, 2 column tiles of 16, 3 K-steps of 4. No guards.
__global__ void gemm1_wmma_kernel(const float* __restrict__ xp, const float* __restrict__ W,
                                  float* __restrict__ out, int n) {
  int gid  = blockIdx.x * blockDim.x + threadIdx.x;
  int wave = gid >> 5;
  int lane = threadIdx.x & 31;
  int row0 = wave * 16;
  if (row0 >= n) return;              // uniform per wave
  int half = lane >> 4, l = lane & 15;

  v8f c0 = {0,0,0,0,0,0,0,0};
  v8f c1 = {0,0,0,0,0,0,0,0};
  const float* xr = xp + ((size_t)(row0 + l) << 4);

#pragma unroll
  for (int kc = 0; kc < 3; ++kc) {
    int k = kc * 4 + half * 2;        // 32-bit A 16x4 layout: lanes 16-31 hold K+2,K+3
    v2f a, b0, b1;
    a.x  = xr[k];
    a.y  = xr[k + 1];
    b0.x = W[k * 32 + l];
    b0.y = W[(k + 1) * 32 + l];
    b1.x = W[k * 32 + 16 + l];
    b1.y = W[(k + 1) * 32 + 16 + l];
    c0 = __builtin_amdgcn_wmma_f32_16x16x4_f32(false, a, false, b0, (short)0, c0, false, false);
    c1 = __builtin_amdgcn_wmma_f32_16x16x4_f32(false, a, false, b1, (short)0, c1, false, false);
  }

#pragma unroll
  for (int r = 0; r < 8; ++r) {       // C/D: VGPR r -> M = half*8 + r, N = l
    int m = row0 + half * 8 + r;
    out[(size_t)m * 32 + l]      = c0[r];
    out[(size_t)m * 32 + 16 + l] = c1[r];
  }
}

// ---------------- dense GEMM 2: xw2 = h1 @ W2  ([N,32] @ [32,64]) ----------------
__global__ void gemm2_wmma_kernel(const float* __restrict__ h, const float* __restrict__ W,
                                  float* __restrict__ out, int n) {
  int gid  = blockIdx.x * blockDim.x + threadIdx.x;
  int wave = gid >> 5;
  int lane = threadIdx.x & 31;
  int row0 = wave * 16;
  if (row0 >= n) return;
  int half = lane >> 4, l = lane & 15;

  v8f c[4];
#pragma unroll
  for (int t = 0; t < 4; ++t) c[t] = (v8f){0,0,0,0,0,0,0,0};

  const float* hr = h + (size_t)(row0 + l) * 32;

#pragma unroll
  for (int kc = 0; kc < 8; ++kc) {
    int k = kc * 4 + half * 2;
    v2f a;
    a.x = hr[k];
    a.y = hr[k + 1];
#pragma unroll
    for (int t = 0; t < 4; ++t) {
      v2f b;
      b.x = W[k * 64 + t * 16 + l];
      b.y = W[(k + 1) * 64 + t * 16 + l];
      c[t] = __builtin_amdgcn_wmma_f32_16x16x4_f32(false, a, false, b, (short)0, c[t], false, false);
    }
  }

#pragma unroll
  for (int t = 0; t < 4; ++t)
#pragma unroll
    for (int r = 0; r < 8; ++r)
      out[(size_t)(row0 + half * 8 + r) * 64 + t * 16 + l] = c[t][r];
}

// ---------------- edge aggregation (scatter-add), feature-group parallel ----------------
// 8 (or 16) lanes per edge, 4 feats each: edge row gather is one coalesced burst,
// atomics land on consecutive addresses in L2.
__global__ void agg32_kernel(const float* __restrict__ xw, const float* __restrict__ dinv,
                             const long long* __restrict__ row, const long long* __restrict__ col,
                             float* __restrict__ agg, int E) {
  long long tid = (long long)blockIdx.x * blockDim.x + threadIdx.x;
  if (tid >= (long long)E * 8) return;
  int e  = (int)(tid >> 3);
  int fg = (int)(tid & 7);
  int r = (int)row[e], c = (int)col[e];
  float nrm = dinv[r] * dinv[c];
  float4 v = *(const float4*)(xw + (size_t)r * 32 + fg * 4);
  float* dst = agg + (size_t)c * 32 + fg * 4;
  atomicAdd(dst + 0, v.x * nrm);
  atomicAdd(dst + 1, v.y * nrm);
  atomicAdd(dst + 2, v.z * nrm);
  atomicAdd(dst + 3, v.w * nrm);
}

__global__ void agg64_kernel(const float* __restrict__ xw, const float* __restrict__ dinv,
                             const long long* __restrict__ row, const long long* __restrict__ col,
                             float* __restrict__ agg, int E) {
  long long tid = (long long)blockIdx.x * blockDim.x + threadIdx.x;
  if (tid >= (long long)E * 16) return;
  int e  = (int)(tid >> 4);
  int fg = (int)(tid & 15);
  int r = (int)row[e], c = (int)col[e];
  float nrm = dinv[r] * dinv[c];
  float4 v = *(const float4*)(xw + (size_t)r * 64 + fg * 4);
  float* dst = agg + (size_t)c * 64 + fg * 4;
  atomicAdd(dst + 0, v.x * nrm);
  atomicAdd(dst + 1, v.y * nrm);
  atomicAdd(dst + 2, v.z * nrm);
  atomicAdd(dst + 3, v.w * nrm);
}

// ---------------- finalize: self-loop + bias + leaky relu (in place over agg) ----------------
// nf is 32 or 64 -> pass log2(nf) to keep index math shift/mask only.
__global__ void fin_kernel(float* agg, const float* __restrict__ xw,
                           const float* __restrict__ dinv, const float* __restrict__ bias,
                           int n, int fshift) {
  long long tid = (long long)blockIdx.x * blockDim.x + threadIdx.x;
  if (tid >= ((long long)n << fshift)) return;
  int node = (int)(tid >> fshift);
  int f    = (int)(tid & ((1 << fshift) - 1));
  float di = dinv[node];
  float v  = agg[tid] + xw[tid] * di * di + bias[f];
  agg[tid] = v > 0.0f ? v : NEG_SLOPE * v;
}

// ---------------- head: sigmoid(cat(x, h2) @ Wfc + bfc) ----------------
__global__ void head_kernel(const float* __restrict__ x, const float* __restrict__ h2,
                            const float* __restrict__ Wfc, const float* __restrict__ bfc,
                            float* __restrict__ out, int n) {
  int i = blockIdx.x * blockDim.x + threadIdx.x;
  if (i >= n) return;
  float z = bfc[0];
  const float* xr = x + (size_t)i * 10;
#pragma unroll
  for (int f = 0; f < 10; ++f) z += xr[f] * Wfc[f];
  const float* hr = h2 + (size_t)i * 64;
#pragma unroll
  for (int f = 0; f < 64; ++f) z += hr[f] * Wfc[10 + f];
  out[i] = 1.0f / (1.0f + __expf(-z));
}

extern "C" void kernel_launch(void* const* d_in, const int* in_sizes, int n_in,
                              void* d_out, int out_size, void* d_ws, size_t ws_size,
                              hipStream_t stream) {
  const float*     x   = (const float*)d_in[0];
  const long long* ei  = (const long long*)d_in[1];  // int64 per reference
  const float*     W1  = (const float*)d_in[2];
  const float*     b1  = (const float*)d_in[3];
  const float*     W2  = (const float*)d_in[4];
  const float*     b2  = (const float*)d_in[5];
  const float*     Wfc = (const float*)d_in[6];
  const float*     bfc = (const float*)d_in[7];
  float* out = (float*)d_out;

  const int N = in_sizes[0] / 10;
  const int E = in_sizes[1] / 2;
  const long long* row = ei;       // sources
  const long long* col = ei + E;   // targets

  // workspace layout (all fp32):
  //   dinv[N] | xpad[N*16] | xw1[N*32] | agg1(->h1)[N*32] | xw2[N*64] | agg2(->h2)[N*64] | W1p[384]
  float* ws   = (float*)d_ws;
  float* dinv = ws;
  float* xpad = dinv + N;
  float* xw1  = xpad + (size_t)N * 16;
  float* agg1 = xw1 + (size_t)N * 32;
  float* xw2  = agg1 + (size_t)N * 32;
  float* agg2 = xw2 + (size_t)N * 64;
  float* W1p  = agg2 + (size_t)N * 64;

  hipMemsetAsync(agg1, 0, (size_t)N * 32 * sizeof(float), stream);
  hipMemsetAsync(agg2, 0, (size_t)N * 64 * sizeof(float), stream);

  // degree -> dinv (in place)
  deg_init_kernel<<<(N + TPB - 1) / TPB, TPB, 0, stream>>>(dinv, N);
  deg_count_kernel<<<(E + TPB - 1) / TPB, TPB, 0, stream>>>(col, dinv, E);
  dinv_kernel<<<(N + TPB - 1) / TPB, TPB, 0, stream>>>(dinv, N);

  // padding for branch-free WMMA GEMM1
  pad_x_kernel<<<(N * 16 + TPB - 1) / TPB, TPB, 0, stream>>>(x, xpad, N);
  pad_w1_kernel<<<2, TPB, 0, stream>>>(W1, W1p);

  const int waves = (N + 15) / 16;
  const int gblk  = (waves * 32 + TPB - 1) / TPB;

  // layer 1
  gemm1_wmma_kernel<<<gblk, TPB, 0, stream>>>(xpad, W1p, xw1, N);
  {
    long long t = (long long)E * 8;
    agg32_kernel<<<(int)((t + TPB - 1) / TPB), TPB, 0, stream>>>(xw1, dinv, row, col, agg1, E);
  }
  fin_kernel<<<(int)(((long long)N * 32 + TPB - 1) / TPB), TPB, 0, stream>>>(agg1, xw1, dinv, b1, N, 5);

  // layer 2 (agg1 now holds h1)
  gemm2_wmma_kernel<<<gblk, TPB, 0, stream>>>(agg1, W2, xw2, N);
  {
    long long t = (long long)E * 16;
    agg64_kernel<<<(int)((t + TPB - 1) / TPB), TPB, 0, stream>>>(xw2, dinv, row, col, agg2, E);
  }
  fin_kernel<<<(int)(((long long)N * 64 + TPB - 1) / TPB), TPB, 0, stream>>>(agg2, xw2, dinv, b2, N, 6);

  // head (agg2 now holds h2)
  head_kernel<<<(N + TPB - 1) / TPB, TPB, 0, stream>>>(x, agg2, Wfc, bfc, out, N);
}